// SparseLinear_19713899889439
// MI455X (gfx1250) — compile-verified
//
#include <hip/hip_runtime.h>

typedef float v2f __attribute__((ext_vector_type(2)));
typedef float v8f __attribute__((ext_vector_type(8)));

#define NUMX 500000
#define DD   768
#define BB   256
#define SS   1000
#define SPLIT 4
#define S_PER_BLOCK 256            /* sBase stays 16B-aligned in the output */
#define GROUPS 16                  /* 256 / 16 */

__global__ __launch_bounds__(256)
void sparse_linear_wmma(const float* __restrict__ embs,
                        const int*   __restrict__ shortlist,
                        const float* __restrict__ weight,
                        const float* __restrict__ bias,
                        float*       __restrict__ out) {
    __shared__ __align__(16) float sE[DD];
    const int b = blockIdx.x;
    const int t = threadIdx.x;

    for (int i = t; i < DD; i += 256) sE[i] = embs[b * DD + i];
    __syncthreads();

    const int lane  = t & 31;
    const int wave  = t >> 5;
    const int row16 = lane & 15;          // which of the 16 gathered rows this lane serves
    const int off4  = (lane >> 4) << 2;   // element offset within 8-wide chunk: 0 or 4

    // B operand: only column N=0 of D is ever read, so B columns 1..15 are
    // don't-care. Every lane streams the embedding with the SAME uniform
    // stride -> single DS address add per unrolled trip, immediate offsets.
    const float* ePtr = sE + off4;

    const int sBase  = blockIdx.y * S_PER_BLOCK;
    const int sLimit = min(SS, sBase + S_PER_BLOCK);

    for (int g = wave; g < GROUPS; g += 8) {
        const int s0 = sBase + g * 16;
        if (s0 >= sLimit) break;                    // uniform per wave: empty trailing group

        int sg    = s0 + row16;
        int sSafe = (sg < sLimit) ? sg : (sLimit - 1);
        int idx   = shortlist[b * SS + sSafe];
        idx = (idx < 0 || idx >= NUMX) ? 0 : idx;   // safety clamp (no OOB gathers)

        // K-slot remap: lanes 0-15 stream elements [8c..8c+3] of their row,
        // lanes 16-31 stream [8c+4..8c+7]; two WMMAs per float4 consume (x,y),(z,w).
        const float* rowPtr = weight + (size_t)idx * DD + off4;
        const float  myBias = bias[idx];

        // Accumulator C: bias pre-loaded into column N=0.
        // C/D layout: VGPR j -> lane 0 = (M=j,N=0), lane 16 = (M=j+8,N=0).
        v8f c;
#pragma unroll
        for (int j = 0; j < 8; ++j) {
            float bj0 = __shfl(myBias, j, 32);
            float bj1 = __shfl(myBias, j + 8, 32);
            c[j] = (lane == 0) ? bj0 : ((lane == 16) ? bj1 : 0.0f);
        }

        const float* ep = ePtr;
#pragma unroll 4
        for (int ch = 0; ch < DD / 8; ++ch) {
            const float4 av = *(const float4*)(rowPtr);   // 16B gather, 512B/wave-instr
            const float4 ev = *(const float4*)(ep);       // ds_load_b128, uniform stride
            rowPtr += 8;
            ep     += 8;

            v2f a0; a0[0] = av.x; a0[1] = av.y;
            v2f b0; b0[0] = ev.x; b0[1] = ev.y;
            c = __builtin_amdgcn_wmma_f32_16x16x4_f32(false, a0, false, b0,
                                                      (short)0, c, false, false);
            v2f a1; a1[0] = av.z; a1[1] = av.w;
            v2f b1; b1[0] = ev.z; b1[1] = ev.w;
            c = __builtin_amdgcn_wmma_f32_16x16x4_f32(false, a1, false, b1,
                                                      (short)0, c, false, false);
        }

        // Column-0 extraction: lane 0 holds rows 0..7, lane 16 holds rows 8..15.
        if (s0 + 16 <= sLimit) {            // full group: vectorized 16B-aligned stores
            if (lane == 0 || lane == 16) {
                float* op = out + b * SS + s0 + ((lane == 16) ? 8 : 0);
                *(float4*)(op)     = make_float4(c[0], c[1], c[2], c[3]);
                *(float4*)(op + 4) = make_float4(c[4], c[5], c[6], c[7]);
            }
        } else {                            // ragged tail: guarded scalar stores
            if (lane == 0 || lane == 16) {
                const int mBase = (lane == 0) ? 0 : 8;
#pragma unroll
                for (int j = 0; j < 8; ++j) {
                    int s = s0 + mBase + j;
                    if (s < sLimit) out[b * SS + s] = c[j];
                }
            }
        }
    }
}

extern "C" void kernel_launch(void* const* d_in, const int* in_sizes, int n_in,
                              void* d_out, int out_size, void* d_ws, size_t ws_size,
                              hipStream_t stream) {
    (void)in_sizes; (void)n_in; (void)d_ws; (void)ws_size; (void)out_size;
    const float* embs      = (const float*)d_in[0];
    const int*   shortlist = (const int*)  d_in[1];
    const float* weight    = (const float*)d_in[2];
    const float* bias      = (const float*)d_in[3];
    float*       out       = (float*)d_out;

    dim3 grid(BB, SPLIT);
    sparse_linear_wmma<<<grid, 256, 0, stream>>>(embs, shortlist, weight, bias, out);
}